// SG_15857019257346
// MI455X (gfx1250) — compile-verified
//
#include <hip/hip_runtime.h>
#include <math.h>

// Problem constants (from reference): B=8, N=8192, T=512, E=64, topk=2
#define Bdim 8
#define Ndim 8192
#define Tdim 512
#define Edim 64
#define TOKENS (Bdim * Ndim)               // 65536 tokens
#define INV_TEMP 0.1f                      // 1/TEMP
#define INV_SIG_SQRT2 7.0710678118654755f  // 1/(0.1*sqrt(2))

#define WSTRIDE 516   // padded LDS row stride for W (dwords)
#define LSTRIDE 65    // padded LDS row stride for logits (dwords)

typedef __attribute__((ext_vector_type(2))) float v2f;
typedef __attribute__((ext_vector_type(8))) float v8f;

__device__ __forceinline__ bool pair_gt(float av, int ai, float bv, int bi) {
    // descending by value; tie -> lower index first (matches jax top_k)
    return (av > bv) || (av == bv && ai < bi);
}

extern __shared__ float smem[];

// Fused epilogue for one 16-token tile: scatter WMMA C fragments to LDS rows,
// then per-token softmax / top-2 / normalization / smooth erf gate.
// Called uniformly by all threads (contains barriers).
__device__ __forceinline__ void epilogue_tile(
    float* __restrict__ sL, v8f a0, v8f a1, v8f a2, v8f a3,
    int lane, int row, int kh, long gtok0, float be0, float be1,
    float* __restrict__ out_ew, float* __restrict__ out_idx,
    float* __restrict__ out_nw, float* __restrict__ out_sp)
{
    // C/D layout: VGPR v, lanes 0-15 -> M=v, N=lane; lanes 16-31 -> M=v+8.
    #pragma unroll
    for (int v = 0; v < 8; ++v) {
        int m = v + 8 * kh;
        float* p = sL + m * LSTRIDE;
        p[ 0 + row] = a0[v];
        p[16 + row] = a1[v];
        p[32 + row] = a2[v];
        p[48 + row] = a3[v];
    }
    __syncthreads();

    for (int t = 0; t < 16; ++t) {
        // each lane owns experts {lane, lane+32} for this token
        float l0 = (sL[t * LSTRIDE + lane     ] + be0) * INV_TEMP;
        float l1 = (sL[t * LSTRIDE + lane + 32] + be1) * INV_TEMP;

        float m = fmaxf(l0, l1);
        #pragma unroll
        for (int off = 16; off > 0; off >>= 1)
            m = fmaxf(m, __shfl_xor(m, off, 32));

        float e0 = expf(l0 - m), e1 = expf(l1 - m);
        float s = e0 + e1;
        #pragma unroll
        for (int off = 16; off > 0; off >>= 1)
            s += __shfl_xor(s, off, 32);
        float inv_s = 1.0f / s;
        float w0 = e0 * inv_s, w1 = e1 * inv_s;

        // local top-2 (value, index)
        float v0, v1f; int i0, i1;
        if (pair_gt(w0, lane, w1, lane + 32)) { v0 = w0; i0 = lane;      v1f = w1; i1 = lane + 32; }
        else                                  { v0 = w1; i0 = lane + 32; v1f = w0; i1 = lane; }

        // wave-wide top-2 merge
        #pragma unroll
        for (int off = 16; off > 0; off >>= 1) {
            float ov0 = __shfl_xor(v0,  off, 32);
            int   oi0 = __shfl_xor(i0,  off, 32);
            float ov1 = __shfl_xor(v1f, off, 32);
            int   oi1 = __shfl_xor(i1,  off, 32);
            bool aF  = pair_gt(v0, i0, ov0, oi0);
            float nv0 = aF ? v0  : ov0;  int ni0 = aF ? i0  : oi0;
            float c1v = aF ? ov0 : v0;   int c1i = aF ? oi0 : i0;
            float c2v = aF ? v1f : ov1;  int c2i = aF ? i1  : oi1;
            bool cF  = pair_gt(c1v, c1i, c2v, c2i);
            v0 = nv0; i0 = ni0;
            v1f = cF ? c1v : c2v; i1 = cF ? c1i : c2i;
        }

        float kth = v1f;   // 2nd-largest softmax weight
        long  g   = gtok0 + t;

        out_ew[g * Edim + lane]      = w0;
        out_ew[g * Edim + lane + 32] = w1;
        out_sp[g * Edim + lane]      = 0.5f * (1.0f + erff((l0 - kth) * INV_SIG_SQRT2));
        out_sp[g * Edim + lane + 32] = 0.5f * (1.0f + erff((l1 - kth) * INV_SIG_SQRT2));
        if (lane == 0) {
            float denom = 1.0f / (v0 + v1f + 1e-12f);
            out_idx[g * 2 + 0] = (float)i0;
            out_idx[g * 2 + 1] = (float)i1;
            out_nw [g * 2 + 0] = v0  * denom;
            out_nw [g * 2 + 1] = v1f * denom;
        }
    }
    __syncthreads();   // region reused by caller's next tile
}

// One block = 128 threads = 4 waves. Each wave computes TWO 16-token x
// 64-expert tiles with V_WMMA_F32_16X16X4_F32, sharing each B (W) fragment
// across both A tiles (halves LDS traffic per WMMA). K loop over T=512.
__global__ __launch_bounds__(128) void moe_router_kernel(
    const float* __restrict__ x,     // [B*N, T]
    const float* __restrict__ W,     // [E, T]
    const float* __restrict__ bias,  // [E]
    float* __restrict__ out)         // ew | idx | nw | smooth (flat, f32)
{
    const int tid = threadIdx.x;

    // ---- stage W (E x T = 128KB) into LDS with padded rows ----
    for (int i = tid; i < Edim * Tdim; i += 128) {
        int e = i >> 9;              // /512
        int t = i & (Tdim - 1);
        smem[e * WSTRIDE + t] = W[i];
    }
    __syncthreads();

    const int wave = tid >> 5;
    const int lane = tid & 31;
    const int row  = lane & 15;      // M (for A) / N (for B,C,D)
    const int kh   = lane >> 4;      // K-half selector

    const long tok0 = (long)blockIdx.x * 128 + (long)wave * 32;
    const float* xrow0 = x + (tok0 + row) * Tdim + 2 * kh;        // tile 0
    const float* xrow1 = xrow0 + 16 * Tdim;                       // tile 1
    const float* sW    = smem;
    const int    boff  = row * WSTRIDE + 2 * kh;

    // ---- GEMM: 2 token tiles x 4 expert tiles, K=512 in steps of 4 ----
    v8f p00 = {}, p01 = {}, p02 = {}, p03 = {};   // tile0 x experts 0..63
    v8f p10 = {}, p11 = {}, p12 = {}, p13 = {};   // tile1 x experts 0..63
    #pragma unroll 2
    for (int k = 0; k < Tdim; k += 4) {
        v2f a0 = *(const v2f*)(xrow0 + k);
        v2f a1 = *(const v2f*)(xrow1 + k);
        v2f b0 = *(const v2f*)(sW + boff + 0 * 16 * WSTRIDE + k);
        v2f b1 = *(const v2f*)(sW + boff + 1 * 16 * WSTRIDE + k);
        v2f b2 = *(const v2f*)(sW + boff + 2 * 16 * WSTRIDE + k);
        v2f b3 = *(const v2f*)(sW + boff + 3 * 16 * WSTRIDE + k);
        p00 = __builtin_amdgcn_wmma_f32_16x16x4_f32(false, a0, false, b0, (short)0, p00, false, false);
        p10 = __builtin_amdgcn_wmma_f32_16x16x4_f32(false, a1, false, b0, (short)0, p10, false, false);
        p01 = __builtin_amdgcn_wmma_f32_16x16x4_f32(false, a0, false, b1, (short)0, p01, false, false);
        p11 = __builtin_amdgcn_wmma_f32_16x16x4_f32(false, a1, false, b1, (short)0, p11, false, false);
        p02 = __builtin_amdgcn_wmma_f32_16x16x4_f32(false, a0, false, b2, (short)0, p02, false, false);
        p12 = __builtin_amdgcn_wmma_f32_16x16x4_f32(false, a1, false, b2, (short)0, p12, false, false);
        p03 = __builtin_amdgcn_wmma_f32_16x16x4_f32(false, a0, false, b3, (short)0, p03, false, false);
        p13 = __builtin_amdgcn_wmma_f32_16x16x4_f32(false, a1, false, b3, (short)0, p13, false, false);
        __builtin_prefetch(xrow0 + k + 128, 0, 0);
        __builtin_prefetch(xrow1 + k + 128, 0, 0);
    }

    // ---- fused epilogue (two tiles through the same per-wave scratch) ----
    float* sL = smem + Edim * WSTRIDE + wave * 16 * LSTRIDE;
    const float be0 = bias[lane];
    const float be1 = bias[lane + 32];
    float* out_ew  = out;
    float* out_idx = out + (long)TOKENS * Edim;
    float* out_nw  = out + (long)TOKENS * Edim + (long)TOKENS * 2;
    float* out_sp  = out + (long)TOKENS * Edim + (long)TOKENS * 4;

    epilogue_tile(sL, p00, p01, p02, p03, lane, row, kh, tok0,
                  be0, be1, out_ew, out_idx, out_nw, out_sp);
    epilogue_tile(sL, p10, p11, p12, p13, lane, row, kh, tok0 + 16,
                  be0, be1, out_ew, out_idx, out_nw, out_sp);
}

extern "C" void kernel_launch(void* const* d_in, const int* in_sizes, int n_in,
                              void* d_out, int out_size, void* d_ws, size_t ws_size,
                              hipStream_t stream) {
    (void)in_sizes; (void)n_in; (void)d_ws; (void)ws_size; (void)out_size;
    const float* x    = (const float*)d_in[0];
    const float* W    = (const float*)d_in[1];
    const float* bvec = (const float*)d_in[2];
    // d_in[3] is topk (==2); hardcoded in kernel for this problem size.
    float* out = (float*)d_out;

    dim3 grid(TOKENS / 128);  // 512 blocks; 4 waves x 32 tokens each
    dim3 block(128);
    size_t shmem = (size_t)(Edim * WSTRIDE + 4 * 16 * LSTRIDE) * sizeof(float);
    hipLaunchKernelGGL(moe_router_kernel, grid, block, shmem, stream,
                       x, W, bvec, out);
}